// NodeDenoisingADMM_50723563766555
// MI455X (gfx1250) — compile-verified
//
#include <hip/hip_runtime.h>
#include <cstdint>

// ---------------- problem constants (from reference) ----------------
constexpr int   NND   = 50000;       // nodes
constexpr int   FT    = 128;         // features
constexpr int   FT4   = FT / 4;      // float4 per row = 32 = one wave
constexpr int   NNZE  = 800000;      // nnz per operator
constexpr int   KOPS  = 3;           // operators
constexpr int   ITERS = 14;
// GAMMA == 1.0 is hardcoded into the math below.

constexpr size_t NF   = (size_t)NND * FT;       // 6.4M floats
constexpr size_t KNF  = (size_t)KOPS * NF;      // 19.2M floats

#define CDIV(a, b) (((a) + (b) - 1) / (b))

// ---------------- CDNA5 async-to-LDS staging (probe-confirmed present) ----------------
#if defined(__has_builtin)
#if __has_builtin(__builtin_amdgcn_global_load_async_to_lds_b32)
#define HAVE_ASYNC_LDS 1
#endif
#endif
#ifndef HAVE_ASYNC_LDS
#define HAVE_ASYNC_LDS 0
#endif

typedef __attribute__((address_space(1))) int gint_t;   // global int*
typedef __attribute__((address_space(3))) int lint_t;   // LDS int*

__device__ __forceinline__ void async_b32(const void* g, const void* l) {
#if HAVE_ASYNC_LDS
  gint_t* gp = (gint_t*)(unsigned long long)g;
  // generic LDS pointer carries the LDS byte offset in its low 32 bits
  lint_t* lp = (lint_t*)(unsigned)(unsigned long long)l;
  __builtin_amdgcn_global_load_async_to_lds_b32(gp, lp, 0, 0);
#endif
}

// Stage one (col,val) pair per lane into wave-private LDS slots (2 async ops).
__device__ __forceinline__ void stage_cv(const unsigned* gc, const float* gv,
                                         unsigned* lc, float* lv) {
#if HAVE_ASYNC_LDS
  async_b32((const void*)gc, (const void*)lc);
  async_b32((const void*)gv, (const void*)lv);
#else
  *lc = *gc;
  *lv = *gv;
#endif
}

// Wait until at most the newest stage's 2 async ops are outstanding
// (in-order completion => the previous chunk has fully landed in LDS).
__device__ __forceinline__ void wait_async_le2() {
#if HAVE_ASYNC_LDS
#if __has_builtin(__builtin_amdgcn_s_wait_asynccnt)
  __builtin_amdgcn_s_wait_asynccnt(2);
#else
  asm volatile("s_wait_asynccnt 2" ::: "memory");
#endif
  asm volatile("" ::: "memory");
#elif __has_builtin(__builtin_amdgcn_wave_barrier)
  __builtin_amdgcn_wave_barrier();
#endif
}

__device__ __forceinline__ void wait_async_le0() {
#if HAVE_ASYNC_LDS
#if __has_builtin(__builtin_amdgcn_s_wait_asynccnt)
  __builtin_amdgcn_s_wait_asynccnt(0);
#else
  asm volatile("s_wait_asynccnt 0" ::: "memory");
#endif
  asm volatile("" ::: "memory");
#elif __has_builtin(__builtin_amdgcn_wave_barrier)
  __builtin_amdgcn_wave_barrier();
#endif
}

__device__ __forceinline__ unsigned umin32(unsigned a, unsigned b) { return a < b ? a : b; }

// ---------------- CSR build (edges static across all 28 SpMM passes) ----------------
__global__ void k_zero_u32(unsigned* p, int n) {
  int t = blockIdx.x * blockDim.x + threadIdx.x;
  if (t < n) p[t] = 0u;
}

__global__ void k_hist(const int* __restrict__ rows, unsigned* __restrict__ cnt) {
  int t = blockIdx.x * blockDim.x + threadIdx.x;
  if (t >= KOPS * NNZE) return;
  int k = t / NNZE;
  int r = rows[t];
  atomicAdd(&cnt[(size_t)k * NND + r], 1u);
}

// Single-block exclusive scan per operator: cnt[k][0..N-1] -> rowptr[k][0..N]
__global__ void k_scan(const unsigned* __restrict__ cnt, unsigned* __restrict__ rowptr) {
  __shared__ unsigned sm[1024];
  for (int k = 0; k < KOPS; ++k) {
    const unsigned* c = cnt + (size_t)k * NND;
    unsigned* rp = rowptr + (size_t)k * (NND + 1);
    unsigned carry = 0;
    for (int base = 0; base < NND; base += 1024) {
      int i = base + (int)threadIdx.x;
      unsigned v = (i < NND) ? c[i] : 0u;
      sm[threadIdx.x] = v;
      __syncthreads();
      unsigned run = v;
      for (int off = 1; off < 1024; off <<= 1) {
        unsigned t = (threadIdx.x >= (unsigned)off) ? sm[threadIdx.x - off] : 0u;
        __syncthreads();
        run += t;
        sm[threadIdx.x] = run;
        __syncthreads();
      }
      if (i < NND) rp[i] = carry + run - v;  // exclusive
      unsigned chunk_total = sm[1023];
      __syncthreads();
      carry += chunk_total;
    }
    if (threadIdx.x == 0) rp[NND] = carry;
    __syncthreads();
  }
}

__global__ void k_init_cursor(const unsigned* __restrict__ rowptr, unsigned* __restrict__ cur) {
  int t = blockIdx.x * blockDim.x + threadIdx.x;
  if (t >= KOPS * NND) return;
  int k = t / NND, r = t - k * NND;
  cur[t] = rowptr[(size_t)k * (NND + 1) + r];
}

__global__ void k_scatter(const int* __restrict__ rows, const int* __restrict__ cols,
                          const float* __restrict__ vals, unsigned* __restrict__ cur,
                          unsigned* __restrict__ cols_s, float* __restrict__ vals_s) {
  int t = blockIdx.x * blockDim.x + threadIdx.x;
  if (t >= KOPS * NNZE) return;
  int k = t / NNZE;
  int r = rows[t];
  unsigned pos = atomicAdd(&cur[(size_t)k * NND + r], 1u);
  cols_s[(size_t)k * NNZE + pos] = (unsigned)cols[t];
  vals_s[(size_t)k * NNZE + pos] = vals[t];
}

// ---------------- shared SpMM row-accumulate core ----------------
// One wave per row, one float4 per lane (FT4 == 32).
// Double-buffered async-to-LDS staging of (col,val) chunks:
//   stage chunk i+1, wait ASYNCcnt<=2 (chunk i landed, i+1 in flight), consume chunk i.
__device__ __forceinline__ float4 row_accum(unsigned start, unsigned end,
                                            const unsigned* __restrict__ gc,
                                            const float* __restrict__ gv,
                                            const float* __restrict__ X,
                                            unsigned* s_cols, float* s_vals,
                                            int sb2 /* wave*64 */, int lane, float4 acc) {
  if (start >= end) return acc;
  int cur = 0;
  {  // prologue: stage first chunk
    unsigned n = umin32(32u, end - start);
    unsigned idx = start + ((unsigned)lane < n ? (unsigned)lane : (n - 1));
    stage_cv(gc + idx, gv + idx, &s_cols[sb2 + lane], &s_vals[sb2 + lane]);
  }
  for (unsigned s = start; s < end; s += 32, cur ^= 1) {
    unsigned n  = umin32(32u, end - s);
    unsigned ns = s + 32;
    if (ns < end) {  // stage ahead into the other buffer, then partial wait
      unsigned nn  = umin32(32u, end - ns);
      unsigned idx = ns + ((unsigned)lane < nn ? (unsigned)lane : (nn - 1));
      int nb = sb2 + ((cur ^ 1) << 5);
      stage_cv(gc + idx, gv + idx, &s_cols[nb + lane], &s_vals[nb + lane]);
      wait_async_le2();
    } else {
      wait_async_le0();
    }
    int b = sb2 + (cur << 5);
    for (unsigned j = 0; j < n; ++j) {
      unsigned c = s_cols[b + j];
      float vv   = s_vals[b + j];
      if (j + 1 < n) {
        unsigned cn = s_cols[b + j + 1];
        __builtin_prefetch((const void*)(X + (size_t)cn * FT + lane * 4), 0, 1);
      }
      float4 xv = ((const float4*)(X + (size_t)c * FT))[lane];
      acc.x = fmaf(vv, xv.x, acc.x);
      acc.y = fmaf(vv, xv.y, acc.y);
      acc.z = fmaf(vv, xv.z, acc.z);
      acc.w = fmaf(vv, xv.w, acc.w);
    }
  }
  return acc;
}

__device__ __forceinline__ float vstep(float wu, float y, float thr) {
  float x = wu + y;                                       // WU + Y (gamma = 1)
  float z = fmaxf(x - thr, 0.f) - fmaxf(-x - thr, 0.f);   // soft threshold
  return y - z;                                           // v = Y - Z
}

// ---------------- fused kernel 1: acc = W_k @ X ; Y = V_old + acc ; V = Y - soft(acc+Y, thr)
__global__ void k_spmm_v(float* __restrict__ V, const float* __restrict__ X,
                         const float* __restrict__ d,
                         const unsigned* __restrict__ rowptr,
                         const unsigned* __restrict__ cols_s,
                         const float* __restrict__ vals_s, int first) {
  __shared__ unsigned s_cols[8 * 64];
  __shared__ float    s_vals[8 * 64];
  const int wave = threadIdx.x >> 5;
  const int lane = threadIdx.x & 31;
  const int row  = blockIdx.x * 8 + wave;   // NND % 8 == 0: no tail, EXEC all-ones
  const int k    = blockIdx.y;
  const int sb2  = wave * 64;

  const unsigned* rp = rowptr + (size_t)k * (NND + 1);
  float4 acc = row_accum(rp[row], rp[row + 1],
                         cols_s + (size_t)k * NNZE, vals_s + (size_t)k * NNZE,
                         X, s_cols, s_vals, sb2, lane, make_float4(0.f, 0.f, 0.f, 0.f));

  float nu = (k == 0) ? 0.0f : ((k == 1) ? 2.0f : 0.5f);
  float thr = nu * d[row];

  float4* vrow = (float4*)(V + (size_t)k * NF + (size_t)row * FT);
  float4 y;
  if (first) {
    y = make_float4(0.f, 0.f, 0.f, 0.f);          // Y0 == 0, do not read poisoned V
  } else {
    float4 vold = vrow[lane];                     // Y_t = v_{t-1} + W@U_t
    y = make_float4(vold.x + acc.x, vold.y + acc.y, vold.z + acc.z, vold.w + acc.w);
  }
  float4 vnew;
  vnew.x = vstep(acc.x, y.x, thr);
  vnew.y = vstep(acc.y, y.y, thr);
  vnew.z = vstep(acc.z, y.z, thr);
  vnew.w = vstep(acc.w, y.w, thr);
  vrow[lane] = vnew;
}

// ---------------- fused kernel 2: acc = sum_k W_k @ V_k ; U = (d*m^2*F - acc)/(d*m^2+1)
__global__ void k_spmm_u(float* __restrict__ Uout, const float* __restrict__ Vbase,
                         const float* __restrict__ Fm, const float* __restrict__ d,
                         const float* __restrict__ mask,
                         const unsigned* __restrict__ rowptr,
                         const unsigned* __restrict__ cols_s,
                         const float* __restrict__ vals_s) {
  __shared__ unsigned s_cols[8 * 64];
  __shared__ float    s_vals[8 * 64];
  const int wave = threadIdx.x >> 5;
  const int lane = threadIdx.x & 31;
  const int row  = blockIdx.x * 8 + wave;
  const int sb2  = wave * 64;

  float4 acc = make_float4(0.f, 0.f, 0.f, 0.f);
  for (int k = 0; k < KOPS; ++k) {
    const unsigned* rp = rowptr + (size_t)k * (NND + 1);
    acc = row_accum(rp[row], rp[row + 1],
                    cols_s + (size_t)k * NNZE, vals_s + (size_t)k * NNZE,
                    Vbase + (size_t)k * NF, s_cols, s_vals, sb2, lane, acc);
  }

  float m = mask[row];
  float dm = d[row] * m * m;
  float inv = 1.0f / (dm + 1.0f);
  float4 f = ((const float4*)(Fm + (size_t)row * FT))[lane];
  float4 u;
  u.x = (dm * f.x - acc.x) * inv;
  u.y = (dm * f.y - acc.y) * inv;
  u.z = (dm * f.z - acc.z) * inv;
  u.w = (dm * f.w - acc.w) * inv;
  ((float4*)(Uout + (size_t)row * FT))[lane] = u;
}

// ---------------- driver ----------------
extern "C" void kernel_launch(void* const* d_in, const int* in_sizes, int n_in,
                              void* d_out, int out_size, void* d_ws, size_t ws_size,
                              hipStream_t stream) {
  const float* Fm     = (const float*)d_in[0];
  const int*   w_rows = (const int*)d_in[1];
  const int*   w_cols = (const int*)d_in[2];
  const float* w_vals = (const float*)d_in[3];
  const float* dvec   = (const float*)d_in[4];
  const float* mask   = (const float*)d_in[5];

  // workspace carving (256B aligned); total ~123 MB
  char* p = (char*)d_ws;
  auto carve = [&](size_t bytes) -> void* {
    void* r = (void*)p;
    p += (bytes + 255) & ~(size_t)255;
    return r;
  };
  float*    U      = (float*)carve(NF * 4);
  float*    V      = (float*)carve(KNF * 4);
  unsigned* rowptr = (unsigned*)carve((size_t)KOPS * (NND + 1) * 4);
  unsigned* cnt    = (unsigned*)carve((size_t)KOPS * NND * 4);
  unsigned* cur    = (unsigned*)carve((size_t)KOPS * NND * 4);
  unsigned* cols_s = (unsigned*)carve((size_t)KOPS * NNZE * 4);
  float*    vals_s = (float*)carve((size_t)KOPS * NNZE * 4);
  (void)ws_size; (void)n_in; (void)in_sizes; (void)out_size;

  const int B = 256;
  const int edges  = KOPS * NNZE;
  const int rows_g = NND / 8;                  // 6250 blocks, 1 wave per row

  // --- build CSR once per launch ---
  k_zero_u32<<<CDIV(KOPS * NND, B), B, 0, stream>>>(cnt, KOPS * NND);
  k_hist<<<CDIV(edges, B), B, 0, stream>>>(w_rows, cnt);
  k_scan<<<1, 1024, 0, stream>>>(cnt, rowptr);
  k_init_cursor<<<CDIV(KOPS * NND, B), B, 0, stream>>>(rowptr, cur);
  k_scatter<<<CDIV(edges, B), B, 0, stream>>>(w_rows, w_cols, w_vals, cur, cols_s, vals_s);

  // --- ADMM iterations: state carried is V (= v_til); U_0 = F fed directly ---
  for (int it = 0; it < ITERS; ++it) {
    const float* X = (it == 0) ? Fm : U;
    k_spmm_v<<<dim3(rows_g, KOPS), B, 0, stream>>>(V, X, dvec, rowptr, cols_s, vals_s,
                                                   it == 0 ? 1 : 0);
    float* u_dst = (it == ITERS - 1) ? (float*)d_out : U;
    k_spmm_u<<<rows_g, B, 0, stream>>>(u_dst, V, Fm, dvec, mask, rowptr, cols_s, vals_s);
  }
}